// LevelHeteroConvLayer_14087492731175
// MI455X (gfx1250) — compile-verified
//
#include <hip/hip_runtime.h>

typedef __attribute__((ext_vector_type(2))) float v2f;
typedef __attribute__((ext_vector_type(4))) float v4f;
typedef __attribute__((ext_vector_type(8))) float v8f;

#define D 256
#define N_WORD 30000
#define N_TOPIC 512
#define N_DOC 10000
#define E_WT 262144
#define E_WD 262144
#define E_TD 131072

#define WMMA_F32(a, b, c) \
  __builtin_amdgcn_wmma_f32_16x16x4_f32(false, (a), false, (b), (short)0, (c), false, false)

// ---------------------------------------------------------------------------
// GEMM: out[m][n] = sum_k feat[m][k] * W[n][k] + b[n]   (y = x @ W.T + b)
// Dual-weight version. One wave computes a 48(M) x 16(N) slab for BOTH weight
// matrices: 3 A-fragment loads + 2 B-fragment loads feed 6 WMMAs per k-step.
// 30000 = 625 * 48, so no M bounds checks. 8 waves per block.
// ---------------------------------------------------------------------------
__global__ void gemm_dual_wmma(const float* __restrict__ feat,
                               const float* __restrict__ W0, const float* __restrict__ b0,
                               const float* __restrict__ W1, const float* __restrict__ b1,
                               float* __restrict__ out0, float* __restrict__ out1,
                               int M) {
  const int lane  = threadIdx.x & 31;
  const int wave  = threadIdx.x >> 5;
  const int wtile = blockIdx.x * 8 + wave;    // wave-tile index
  const int ntn   = D / 16;                   // 16 N-tiles
  const int mslab = wtile / ntn;              // 48-row slab
  const int nt    = wtile % ntn;
  const int mBase = mslab * 48;
  if (mBase >= M) return;

  const int half  = lane >> 4;                // 0: K pair {0,1}, 1: K pair {2,3}
  const int lr    = lane & 15;
  const int nBase = nt * 16;

  const float* a0p = feat + (size_t)(mBase + lr) * D + 2 * half;
  const float* a1p = a0p + 16 * D;
  const float* a2p = a0p + 32 * D;
  const float* u0p = W0 + (size_t)(nBase + lr) * D + 2 * half;
  const float* u1p = W1 + (size_t)(nBase + lr) * D + 2 * half;

  v8f c00 = {}, c01 = {}, c02 = {};           // W0: m-tiles 0,1,2
  v8f c10 = {}, c11 = {}, c12 = {};           // W1: m-tiles 0,1,2
#pragma unroll 4
  for (int k = 0; k < D; k += 4) {
    v2f a0 = *(const v2f*)(a0p + k);
    v2f a1 = *(const v2f*)(a1p + k);
    v2f a2 = *(const v2f*)(a2p + k);
    v2f u0 = *(const v2f*)(u0p + k);
    v2f u1 = *(const v2f*)(u1p + k);
    c00 = WMMA_F32(a0, u0, c00);
    c10 = WMMA_F32(a0, u1, c10);
    c01 = WMMA_F32(a1, u0, c01);
    c11 = WMMA_F32(a1, u1, c11);
    c02 = WMMA_F32(a2, u0, c02);
    c12 = WMMA_F32(a2, u1, c12);
  }

  const int col = nBase + lr;
  const float bias0 = b0[col];
  const float bias1 = b1[col];
#pragma unroll
  for (int r = 0; r < 8; ++r) {
    const size_t row = (size_t)(mBase + r + 8 * half);  // lanes 16-31 hold M = r+8
    out0[(row     ) * D + col] = c00[r] + bias0;
    out0[(row + 16) * D + col] = c01[r] + bias0;
    out0[(row + 32) * D + col] = c02[r] + bias0;
    out1[(row     ) * D + col] = c10[r] + bias1;
    out1[(row + 16) * D + col] = c11[r] + bias1;
    out1[(row + 32) * D + col] = c12[r] + bias1;
  }
}

// Single-output 16x16-per-wave variant (topic features, tiny M=512).
__global__ void gemm_single_wmma(const float* __restrict__ feat,
                                 const float* __restrict__ W, const float* __restrict__ b,
                                 float* __restrict__ out, int M) {
  const int lane = threadIdx.x & 31;
  const int wave = threadIdx.x >> 5;
  const int tile = blockIdx.x * 8 + wave;
  const int ntn  = D / 16;
  const int mt   = tile / ntn;
  const int nt   = tile % ntn;
  if (mt * 16 >= M) return;

  const int half  = lane >> 4;
  const int lr    = lane & 15;
  const int mBase = mt * 16;
  const int nBase = nt * 16;

  const float* aRow = feat + (size_t)(mBase + lr) * D + 2 * half;
  const float* bRow = W    + (size_t)(nBase + lr) * D + 2 * half;

  v8f acc = {};
#pragma unroll 8
  for (int k = 0; k < D; k += 4) {
    v2f a = *(const v2f*)(aRow + k);
    v2f u = *(const v2f*)(bRow + k);
    acc = WMMA_F32(a, u, acc);
  }

  const int col = nBase + lr;
  const float bias = b[col];
#pragma unroll
  for (int r = 0; r < 8; ++r) {
    const int row = mBase + r + 8 * half;
    out[(size_t)row * D + col] = acc[r] + bias;
  }
}

// ---------------------------------------------------------------------------
// Edge scatter: acc[dst[e]][:] += Wh[src[e]][:] * w[e]
// 64 threads per edge, 4 floats per thread (float4 gather + 4 f32 atomics).
// Prefetch the gathered row a few edges ahead (global_prefetch_b8).
// ---------------------------------------------------------------------------
#define PF_EDGES 8
__global__ void edge_scatter(const float* __restrict__ Wh,
                             const int* __restrict__ src, const int* __restrict__ dst,
                             const float* __restrict__ w,
                             float* __restrict__ acc, int E) {
  const int idx = blockIdx.x * blockDim.x + threadIdx.x;
  const int e = idx >> 6;
  if (e >= E) return;
  const int c = (idx & 63) << 2;

  if (e + PF_EDGES < E) {
    __builtin_prefetch(Wh + (size_t)src[e + PF_EDGES] * D + c, 0, 1);
  }

  const float we = w[e];
  const float* p = Wh  + (size_t)src[e] * D + c;
  float*       q = acc + (size_t)dst[e] * D + c;
  v4f v = *(const v4f*)p;
  unsafeAtomicAdd(q + 0, v.x * we);
  unsafeAtomicAdd(q + 1, v.y * we);
  unsafeAtomicAdd(q + 2, v.z * we);
  unsafeAtomicAdd(q + 3, v.w * we);
}

__global__ void edge_degree(const int* __restrict__ dst, float* __restrict__ deg, int E) {
  const int e = blockIdx.x * blockDim.x + threadIdx.x;
  if (e < E) unsafeAtomicAdd(&deg[dst[e]], 1.0f);
}

// ---------------------------------------------------------------------------
// Finalize: topic = sum/deg ; doc = relu(sum_wd/deg_wd + sum_td/deg_td)
// ---------------------------------------------------------------------------
__global__ void finalize_topic(const float* __restrict__ acc, const float* __restrict__ deg,
                               float* __restrict__ out, int total) {
  const int i = blockIdx.x * blockDim.x + threadIdx.x;
  if (i < total) out[i] = acc[i] / fmaxf(deg[i >> 8], 1.0f);
}

__global__ void finalize_doc(const float* __restrict__ acc_wd, const float* __restrict__ deg_wd,
                             const float* __restrict__ acc_td, const float* __restrict__ deg_td,
                             float* __restrict__ out, int total) {
  const int i = blockIdx.x * blockDim.x + threadIdx.x;
  if (i < total) {
    const int r = i >> 8;
    const float m = acc_wd[i] / fmaxf(deg_wd[r], 1.0f)
                  + acc_td[i] / fmaxf(deg_td[r], 1.0f);
    out[i] = fmaxf(m, 0.0f);
  }
}

// ---------------------------------------------------------------------------
extern "C" void kernel_launch(void* const* d_in, const int* in_sizes, int n_in,
                              void* d_out, int out_size, void* d_ws, size_t ws_size,
                              hipStream_t stream) {
  const float* feat_word  = (const float*)d_in[0];
  const float* feat_topic = (const float*)d_in[1];
  // d_in[2] feat_doc: unused (doc nodes only receive messages)
  const int*   wt_src = (const int*)d_in[3];
  const int*   wt_dst = (const int*)d_in[4];
  const int*   wd_src = (const int*)d_in[5];
  const int*   wd_dst = (const int*)d_in[6];
  const int*   td_src = (const int*)d_in[7];
  const int*   td_dst = (const int*)d_in[8];
  const float* w_wt = (const float*)d_in[9];
  const float* w_wd = (const float*)d_in[10];
  const float* w_td = (const float*)d_in[11];
  const float* Wt  = (const float*)d_in[12];
  const float* bt  = (const float*)d_in[13];
  const float* Wd  = (const float*)d_in[14];
  const float* bd  = (const float*)d_in[15];
  const float* Wtd = (const float*)d_in[16];
  const float* btd = (const float*)d_in[17];

  // ---- workspace layout (floats) ----
  float* ws = (float*)d_ws;
  float* Wh_wt = ws;                                   // 30000*256
  float* Wh_wd = Wh_wt + (size_t)N_WORD * D;           // 30000*256
  float* Wh_td = Wh_wd + (size_t)N_WORD * D;           // 512*256
  // zeroed region starts here (accumulators + degrees, contiguous):
  float* acc_topic = Wh_td + (size_t)N_TOPIC * D;      // 512*256
  float* acc_wd    = acc_topic + (size_t)N_TOPIC * D;  // 10000*256
  float* acc_td    = acc_wd + (size_t)N_DOC * D;       // 10000*256
  float* deg_topic = acc_td + (size_t)N_DOC * D;       // 512
  float* deg_wd    = deg_topic + N_TOPIC;              // 10000
  float* deg_td    = deg_wd + N_DOC;                   // 10000
  const size_t zero_floats = (size_t)N_TOPIC * D + 2ull * N_DOC * D
                           + N_TOPIC + 2ull * N_DOC;
  hipMemsetAsync(acc_topic, 0, zero_floats * sizeof(float), stream);

  // ---- output layout ----
  float* out_word  = (float*)d_out;                            // 30000*256
  float* out_topic = out_word + (size_t)N_WORD * D;            // 512*256
  float* out_doc   = out_topic + (size_t)N_TOPIC * D;          // 10000*256

  // word features pass through unchanged
  hipMemcpyAsync(out_word, feat_word, (size_t)N_WORD * D * sizeof(float),
                 hipMemcpyDeviceToDevice, stream);

  // ---- GEMMs (WMMA fp32) ----
  // word: (30000/48)=625 m-slabs * 16 n-tiles = 10000 wave-tiles / 8 = 1250 blocks
  gemm_dual_wmma<<<(N_WORD / 48) * (D / 16) / 8, 256, 0, stream>>>(
      feat_word, Wt, bt, Wd, bd, Wh_wt, Wh_wd, N_WORD);
  // topic: 32*16 = 512 tiles -> 64 blocks
  gemm_single_wmma<<<(N_TOPIC / 16) * (D / 16) / 8, 256, 0, stream>>>(
      feat_topic, Wtd, btd, Wh_td, N_TOPIC);

  // ---- edge scatter (sum) + degrees ----
  edge_scatter<<<(E_WT * 64) / 256, 256, 0, stream>>>(Wh_wt, wt_src, wt_dst, w_wt, acc_topic, E_WT);
  edge_scatter<<<(E_WD * 64) / 256, 256, 0, stream>>>(Wh_wd, wd_src, wd_dst, w_wd, acc_wd, E_WD);
  edge_scatter<<<(E_TD * 64) / 256, 256, 0, stream>>>(Wh_td, td_src, td_dst, w_td, acc_td, E_TD);

  edge_degree<<<(E_WT + 255) / 256, 256, 0, stream>>>(wt_dst, deg_topic, E_WT);
  edge_degree<<<(E_WD + 255) / 256, 256, 0, stream>>>(wd_dst, deg_wd, E_WD);
  edge_degree<<<(E_TD + 255) / 256, 256, 0, stream>>>(td_dst, deg_td, E_TD);

  // ---- finalize ----
  finalize_topic<<<(N_TOPIC * D) / 256, 256, 0, stream>>>(acc_topic, deg_topic, out_topic, N_TOPIC * D);
  finalize_doc<<<(N_DOC * D) / 256, 256, 0, stream>>>(acc_wd, deg_wd, acc_td, deg_td, out_doc, N_DOC * D);
}